// SceneGraphRefiner_68667937129188
// MI455X (gfx1250) — compile-verified
//
#include <hip/hip_runtime.h>
#include <hip/hip_bf16.h>
#include <stdint.h>

typedef float v2f __attribute__((ext_vector_type(2)));
typedef float v8f __attribute__((ext_vector_type(8)));
typedef int   v4i __attribute__((ext_vector_type(4)));

#define GLOBAL_AS __attribute__((address_space(1)))
#define LDS_AS    __attribute__((address_space(3)))

#define GN    48
#define FIN   9
#define H1    4
#define C1    8
#define HC1   32   // 4 heads * 8
#define E0    768
#define ET    816  // +48 self loops
#define OC    9

// ---------------------------------------------------------------------------
// Kernel 1: both GAT layers + LayerNorm + residual, fused in a single block.
// Tiny problem (48 nodes, 816 edges) -> everything lives in LDS.
// Also zeroes the BN stats accumulators for this launch.
// ---------------------------------------------------------------------------
__global__ __launch_bounds__(256) void k_graph(
    const float* __restrict__ obj, const int* __restrict__ eidx,
    const float* __restrict__ W1, const float* __restrict__ as1,
    const float* __restrict__ ad1, const float* __restrict__ b1,
    const float* __restrict__ W2, const float* __restrict__ as2,
    const float* __restrict__ ad2, const float* __restrict__ b2,
    const float* __restrict__ lng, const float* __restrict__ lnb,
    float* __restrict__ xout, float* __restrict__ stats)
{
  __shared__ float sX[GN * FIN];
  __shared__ float sW1[FIN * HC1];
  __shared__ float sH1[GN * HC1];
  __shared__ float sAS[GN * H1], sAD[GN * H1];
  __shared__ float sEval[ET * H1];
  __shared__ float sOut1[GN * HC1];
  __shared__ float sX1[GN * HC1];
  __shared__ float sW2[HC1 * OC];
  __shared__ float sH2[GN * OC];
  __shared__ float sAS2[GN], sAD2[GN];
  __shared__ float sE2[ET];
  __shared__ float sOut2[GN * OC];
  __shared__ int   sSrc[ET], sTgt[ET];

  const int t = threadIdx.x;

  if (t < 32) stats[t] = 0.f;                         // zero BN sums each launch

  for (int i = t; i < GN * FIN; i += 256)  sX[i]  = obj[i];
  for (int i = t; i < FIN * HC1; i += 256) sW1[i] = W1[i];
  for (int i = t; i < HC1 * OC; i += 256)  sW2[i] = W2[i];
  for (int i = t; i < ET; i += 256) {
    sSrc[i] = (i < E0) ? eidx[i]      : (i - E0);
    sTgt[i] = (i < E0) ? eidx[E0 + i] : (i - E0);
  }
  for (int i = t; i < GN * HC1; i += 256) sOut1[i] = 0.f;
  for (int i = t; i < GN * OC; i += 256)  sOut2[i] = 0.f;
  __syncthreads();

  // h1 = X @ W1  (48x9 @ 9x32)
  for (int i = t; i < GN * HC1; i += 256) {
    int n = i / HC1, j = i % HC1;
    float acc = 0.f;
    #pragma unroll
    for (int k = 0; k < FIN; k++) acc += sX[n * FIN + k] * sW1[k * HC1 + j];
    sH1[i] = acc;
  }
  __syncthreads();

  // alpha_src / alpha_dst per (node, head)
  for (int i = t; i < GN * H1; i += 256) {
    int n = i / H1, h = i % H1;
    float a = 0.f, d = 0.f;
    #pragma unroll
    for (int c = 0; c < C1; c++) {
      float hv = sH1[n * HC1 + h * C1 + c];
      a += hv * as1[h * C1 + c];
      d += hv * ad1[h * C1 + c];
    }
    sAS[i] = a; sAD[i] = d;
  }
  __syncthreads();

  // edge scores, leaky relu
  for (int i = t; i < ET * H1; i += 256) {
    int e = i / H1, h = i % H1;
    float v = sAS[sSrc[e] * H1 + h] + sAD[sTgt[e] * H1 + h];
    sEval[i] = (v > 0.f) ? v : 0.2f * v;
  }
  __syncthreads();

  // per-(target,head) softmax; each thread owns one segment (disjoint edges)
  if (t < GN * H1) {
    int n = t / H1, h = t % H1;
    float m = -3.4e38f;
    for (int e = 0; e < ET; e++)
      if (sTgt[e] == n) { float v = sEval[e * H1 + h]; m = v > m ? v : m; }
    float s = 0.f;
    for (int e = 0; e < ET; e++)
      if (sTgt[e] == n) s += expf(sEval[e * H1 + h] - m);
    float inv = 1.f / (s + 1e-16f);
    for (int e = 0; e < ET; e++)
      if (sTgt[e] == n) sEval[e * H1 + h] = expf(sEval[e * H1 + h] - m) * inv;
  }
  __syncthreads();

  // scatter-aggregate: out1[tgt] += h1[src] * alpha   (LDS float atomics)
  for (int i = t; i < ET * HC1; i += 256) {
    int e = i >> 5, j = i & 31;
    int h = j >> 3;
    atomicAdd(&sOut1[sTgt[e] * HC1 + j], sH1[sSrc[e] * HC1 + j] * sEval[e * H1 + h]);
  }
  __syncthreads();

  // x1 = relu(out1 + b1)
  for (int i = t; i < GN * HC1; i += 256) {
    float v = sOut1[i] + b1[i % HC1];
    sX1[i] = v > 0.f ? v : 0.f;
  }
  __syncthreads();

  // h2 = x1 @ W2  (48x32 @ 32x9)
  for (int i = t; i < GN * OC; i += 256) {
    int n = i / OC, j = i % OC;
    float acc = 0.f;
    #pragma unroll
    for (int k = 0; k < HC1; k++) acc += sX1[n * HC1 + k] * sW2[k * OC + j];
    sH2[i] = acc;
  }
  __syncthreads();

  if (t < GN) {
    float a = 0.f, d = 0.f;
    #pragma unroll
    for (int c = 0; c < OC; c++) {
      float hv = sH2[t * OC + c];
      a += hv * as2[c]; d += hv * ad2[c];
    }
    sAS2[t] = a; sAD2[t] = d;
  }
  __syncthreads();

  for (int e = t; e < ET; e += 256) {
    float v = sAS2[sSrc[e]] + sAD2[sTgt[e]];
    sE2[e] = (v > 0.f) ? v : 0.2f * v;
  }
  __syncthreads();

  if (t < GN) {
    float m = -3.4e38f;
    for (int e = 0; e < ET; e++)
      if (sTgt[e] == t) { float v = sE2[e]; m = v > m ? v : m; }
    float s = 0.f;
    for (int e = 0; e < ET; e++)
      if (sTgt[e] == t) s += expf(sE2[e] - m);
    float inv = 1.f / (s + 1e-16f);
    for (int e = 0; e < ET; e++)
      if (sTgt[e] == t) sE2[e] = expf(sE2[e] - m) * inv;
  }
  __syncthreads();

  for (int i = t; i < ET * OC; i += 256) {
    int e = i / OC, c = i % OC;
    atomicAdd(&sOut2[sTgt[e] * OC + c], sH2[sSrc[e] * OC + c] * sE2[e]);
  }
  __syncthreads();

  // LayerNorm(9) + residual, one node per thread
  if (t < GN) {
    float row[OC]; float mu = 0.f;
    #pragma unroll
    for (int c = 0; c < OC; c++) { row[c] = sOut2[t * OC + c] + b2[c]; mu += row[c]; }
    mu *= (1.f / OC);
    float var = 0.f;
    #pragma unroll
    for (int c = 0; c < OC; c++) { float d = row[c] - mu; var += d * d; }
    var *= (1.f / OC);
    float inv = rsqrtf(var + 1e-5f);
    #pragma unroll
    for (int c = 0; c < OC; c++)
      xout[t * OC + c] = lng[c] * (row[c] - mu) * inv + lnb[c] + sX[t * OC + c];
  }
}

// ---------------------------------------------------------------------------
// Kernel 2: y = relu(x @ lin_W + lin_b), (48x9)@(9x2048), full-f32 WMMA.
// V_WMMA_F32_16X16X4_F32, K padded 9->12 (3 wmma per tile).
// Fragment layout per ISA 7.12.2 (f32 tables).
// ---------------------------------------------------------------------------
__global__ __launch_bounds__(256) void k_lin(
    const float* __restrict__ x, const float* __restrict__ Wl,
    const float* __restrict__ bl, float* __restrict__ y)
{
  int wave = (blockIdx.x << 3) + (threadIdx.x >> 5);  // 0..383
  int lane = threadIdx.x & 31;
  int mt = wave % 3;          // 3 M-tiles of 16 rows (48 rows exactly)
  int nt = wave / 3;          // 128 N-tiles of 16 cols
  int hsel = lane >> 4;       // half-wave selector
  int l16  = lane & 15;
  int row = mt * 16 + l16;
  int col = nt * 16 + l16;

  v8f acc = {};
  #pragma unroll
  for (int kk = 0; kk < 3; kk++) {
    int kb = kk * 4 + hsel * 2;          // this half-wave holds K=kb, kb+1
    v2f a, b;
    a[0] = (kb     < FIN) ? x[row * FIN + kb]     : 0.f;
    a[1] = (kb + 1 < FIN) ? x[row * FIN + kb + 1] : 0.f;
    b[0] = (kb     < FIN) ? Wl[kb * 2048 + col]       : 0.f;
    b[1] = (kb + 1 < FIN) ? Wl[(kb + 1) * 2048 + col] : 0.f;
    acc = __builtin_amdgcn_wmma_f32_16x16x4_f32(
        /*neg_a=*/false, a, /*neg_b=*/false, b,
        /*c_mod=*/(short)0, acc, /*reuse_a=*/false, /*reuse_b=*/false);
  }
  float bo = bl[col];
  #pragma unroll
  for (int r = 0; r < 8; r++) {
    int orow = mt * 16 + r + (hsel << 3);   // VGPR r -> rows r / r+8
    float v = acc[r] + bo;
    y[orow * 2048 + col] = v > 0.f ? v : 0.f;
  }
}

// ---------------------------------------------------------------------------
// Kernel 3: ConvTranspose3d 32->16, k=4, s=2, p=1 : (48,32,4,4,4)->(48,16,8,8,8), +bias, relu
// One block per (n, co); input slab + per-co weights staged in LDS.
// ---------------------------------------------------------------------------
__global__ __launch_bounds__(256) void k_ct1(
    const float* __restrict__ in, const float* __restrict__ W,
    const float* __restrict__ bias, float* __restrict__ out)
{
  int n = blockIdx.x >> 4;
  int co = blockIdx.x & 15;
  __shared__ float sIn[32 * 64];
  __shared__ float sW[32 * 64];
  int t = threadIdx.x;
  for (int i = t; i < 2048; i += 256) sIn[i] = in[n * 2048 + i];
  for (int i = t; i < 2048; i += 256) {
    int ci = i >> 6, k = i & 63;
    sW[i] = W[ci * (16 * 64) + co * 64 + k];
  }
  __syncthreads();
  float bo = bias[co];
  for (int o = t; o < 512; o += 256) {
    int oz = o >> 6, oy = (o >> 3) & 7, ox = o & 7;
    float acc = 0.f;
    for (int kz = (oz + 1) & 1; kz < 4; kz += 2) {
      int iz = (oz + 1 - kz) >> 1; if ((unsigned)iz >= 4u) continue;
      for (int ky = (oy + 1) & 1; ky < 4; ky += 2) {
        int iy = (oy + 1 - ky) >> 1; if ((unsigned)iy >= 4u) continue;
        for (int kx = (ox + 1) & 1; kx < 4; kx += 2) {
          int ix = (ox + 1 - kx) >> 1; if ((unsigned)ix >= 4u) continue;
          int ii = iz * 16 + iy * 4 + ix;
          int kk = kz * 16 + ky * 4 + kx;
          #pragma unroll
          for (int ci = 0; ci < 32; ci++)
            acc += sIn[ci * 64 + ii] * sW[ci * 64 + kk];
        }
      }
    }
    float v = acc + bo;
    out[(n * 16 + co) * 512 + o] = v > 0.f ? v : 0.f;
  }
}

// ---------------------------------------------------------------------------
// Kernel 4: ConvTranspose3d 16->9 : (48,16,8,8,8)->(48,9,16,16,16), +bias.
// Also accumulates per-channel sum / sumsq for training-mode BatchNorm3d.
// ---------------------------------------------------------------------------
__global__ __launch_bounds__(256) void k_ct2(
    const float* __restrict__ in, const float* __restrict__ W,
    const float* __restrict__ bias, float* __restrict__ out,
    float* __restrict__ stats)
{
  int n = blockIdx.x / 9;
  int co = blockIdx.x % 9;
  __shared__ float sIn[16 * 512];
  __shared__ float sW[16 * 64];
  __shared__ float rs[256], rq[256];
  int t = threadIdx.x;
  for (int i = t; i < 16 * 512; i += 256) sIn[i] = in[n * (16 * 512) + i];
  for (int i = t; i < 16 * 64; i += 256) {
    int ci = i >> 6, k = i & 63;
    sW[i] = W[ci * (9 * 64) + co * 64 + k];
  }
  __syncthreads();
  float bo = bias[co];
  float ls = 0.f, lq = 0.f;
  for (int o = t; o < 4096; o += 256) {
    int oz = o >> 8, oy = (o >> 4) & 15, ox = o & 15;
    float acc = 0.f;
    for (int kz = (oz + 1) & 1; kz < 4; kz += 2) {
      int iz = (oz + 1 - kz) >> 1; if ((unsigned)iz >= 8u) continue;
      for (int ky = (oy + 1) & 1; ky < 4; ky += 2) {
        int iy = (oy + 1 - ky) >> 1; if ((unsigned)iy >= 8u) continue;
        for (int kx = (ox + 1) & 1; kx < 4; kx += 2) {
          int ix = (ox + 1 - kx) >> 1; if ((unsigned)ix >= 8u) continue;
          int ii = iz * 64 + iy * 8 + ix;
          int kk = kz * 16 + ky * 4 + kx;
          #pragma unroll
          for (int ci = 0; ci < 16; ci++)
            acc += sIn[ci * 512 + ii] * sW[ci * 64 + kk];
        }
      }
    }
    float v = acc + bo;
    out[(n * 9 + co) * 4096 + o] = v;
    ls += v; lq += v * v;
  }
  rs[t] = ls; rq[t] = lq;
  __syncthreads();
  for (int off = 128; off > 0; off >>= 1) {
    if (t < off) { rs[t] += rs[t + off]; rq[t] += rq[t + off]; }
    __syncthreads();
  }
  if (t == 0) {
    atomicAdd(&stats[co], rs[0]);
    atomicAdd(&stats[16 + co], rq[0]);
  }
}

// ---------------------------------------------------------------------------
// Kernel 5: fold BN into affine, trilinear upsample 16^3 -> 64^3 (half-pixel,
// boundary-renormalized == clamp), out = vol + 0.5 * BN(proj).
// One block per (slab, z-chunk); 16-lane groups each produce a 64-float row
// as coalesced float4 stores. Source slab (16 KB) staged in LDS via the
// gfx1250 async-LDS path when available.
// ---------------------------------------------------------------------------
__global__ __launch_bounds__(256) void k_up(
    const float* __restrict__ h, const float* __restrict__ stats,
    const float* __restrict__ bng, const float* __restrict__ bnb,
    const float* __restrict__ vol, const float* __restrict__ pscale,
    float* __restrict__ outp)
{
  int b = blockIdx.x;
  int zc = b & 3;           // 4 z-chunks of 16 output slices
  int slab = b >> 2;        // (n, c) pair, 0..431
  int c = slab % 9;
  __shared__ __align__(16) float s[4096];
  int t = threadIdx.x;
  const float* src = h + (size_t)slab * 4096;

#if defined(__gfx1250__) && __has_builtin(__builtin_amdgcn_global_load_async_to_lds_b128)
  for (int i = t * 4; i < 4096; i += 1024) {
    __builtin_amdgcn_global_load_async_to_lds_b128(
        (GLOBAL_AS v4i*)(src + i),
        (LDS_AS v4i*)(s + i), 0, 0);
  }
  asm volatile("s_wait_asynccnt 0x0" ::: "memory");
#else
  for (int i = t * 4; i < 4096; i += 1024)
    *(float4*)(s + i) = *(const float4*)(src + i);
#endif
  __syncthreads();

  // BN affine fold (per channel), scaled by proj_scale
  float cnt = 48.f * 4096.f;
  float mean = stats[c] / cnt;
  float var  = stats[16 + c] / cnt - mean * mean;
  float inv  = rsqrtf(var + 1e-5f);
  float ps = pscale[0];
  float S = ps * bng[c] * inv;
  float O = ps * (bnb[c] - bng[c] * inv * mean);

  int grp = t >> 4;          // 16 row-groups per block
  int k = t & 15;            // lane within group -> source column k
  int kl = k > 0 ? k - 1 : 0;
  int kr = k < 15 ? k + 1 : 15;
  size_t base = (size_t)slab * 262144;

  for (int rid = grp; rid < 1024; rid += 16) {
    int z = (zc << 4) + (rid >> 6);
    int y = rid & 63;
    float sz = 0.25f * z - 0.375f;
    float fz = floorf(sz);
    int iz0 = (int)fz; float wz1 = sz - fz; int iz1 = iz0 + 1;
    iz0 = iz0 < 0 ? 0 : iz0;  iz1 = iz1 > 15 ? 15 : iz1;
    float sy = 0.25f * y - 0.375f;
    float fy = floorf(sy);
    int iy0 = (int)fy; float wy1 = sy - fy; int iy1 = iy0 + 1;
    iy0 = iy0 < 0 ? 0 : iy0;  iy1 = iy1 > 15 ? 15 : iy1;
    float wz0 = 1.f - wz1, wy0 = 1.f - wy1;
    float w00 = wz0 * wy0, w01 = wz0 * wy1, w10 = wz1 * wy0, w11 = wz1 * wy1;
    const float* r00 = s + iz0 * 256 + iy0 * 16;
    const float* r01 = s + iz0 * 256 + iy1 * 16;
    const float* r10 = s + iz1 * 256 + iy0 * 16;
    const float* r11 = s + iz1 * 256 + iy1 * 16;
    float vC = w00 * r00[k]  + w01 * r01[k]  + w10 * r10[k]  + w11 * r11[k];
    float vL = w00 * r00[kl] + w01 * r01[kl] + w10 * r10[kl] + w11 * r11[kl];
    float vR = w00 * r00[kr] + w01 * r01[kr] + w10 * r10[kr] + w11 * r11[kr];

    size_t oidx = base + (size_t)z * 4096 + (size_t)y * 64 + (k << 2);
    float4 vv = *(const float4*)(vol + oidx);
    float4 o;
    o.x = vv.x + S * (0.375f * vL + 0.625f * vC) + O;
    o.y = vv.y + S * (0.125f * vL + 0.875f * vC) + O;
    o.z = vv.z + S * (0.875f * vC + 0.125f * vR) + O;
    o.w = vv.w + S * (0.625f * vC + 0.375f * vR) + O;
    *(float4*)(outp + oidx) = o;
  }
}

// ---------------------------------------------------------------------------
extern "C" void kernel_launch(void* const* d_in, const int* in_sizes, int n_in,
                              void* d_out, int out_size, void* d_ws, size_t ws_size,
                              hipStream_t stream) {
  const float* obj   = (const float*)d_in[0];
  const int*   eidx  = (const int*)  d_in[1];
  const float* vol   = (const float*)d_in[2];
  const float* W1    = (const float*)d_in[3];
  const float* as1   = (const float*)d_in[4];
  const float* ad1   = (const float*)d_in[5];
  const float* b1    = (const float*)d_in[6];
  const float* W2    = (const float*)d_in[7];
  const float* as2   = (const float*)d_in[8];
  const float* ad2   = (const float*)d_in[9];
  const float* b2    = (const float*)d_in[10];
  const float* lng   = (const float*)d_in[11];
  const float* lnb   = (const float*)d_in[12];
  const float* psc   = (const float*)d_in[13];
  const float* linW  = (const float*)d_in[14];
  const float* linb  = (const float*)d_in[15];
  const float* c1W   = (const float*)d_in[16];
  const float* c1b   = (const float*)d_in[17];
  const float* c2W   = (const float*)d_in[18];
  const float* c2b   = (const float*)d_in[19];
  const float* bng   = (const float*)d_in[20];
  const float* bnb   = (const float*)d_in[21];
  float* out = (float*)d_out;

  float* ws    = (float*)d_ws;
  float* x     = ws;                         // 432  (pad to 512)
  float* y     = ws + 512;                   // 48*2048
  float* c1o   = y  + 48 * 2048;             // 48*16*512
  float* h2    = c1o + 48 * 16 * 512;        // 48*9*4096
  float* stats = h2 + 48 * 9 * 4096;         // 32

  k_graph<<<1, 256, 0, stream>>>(obj, eidx, W1, as1, ad1, b1,
                                 W2, as2, ad2, b2, lng, lnb, x, stats);
  k_lin  <<<48, 256, 0, stream>>>(x, linW, linb, y);
  k_ct1  <<<48 * 16, 256, 0, stream>>>(y, c1W, c1b, c1o);
  k_ct2  <<<48 * 9, 256, 0, stream>>>(c1o, c2W, c2b, h2, stats);
  k_up   <<<48 * 9 * 4, 256, 0, stream>>>(h2, stats, bng, bnb, vol, psc, out);
}